// G2_51539607552123
// MI455X (gfx1250) — compile-verified
//
#include <hip/hip_runtime.h>
#include <hip/hip_bf16.h>

typedef __attribute__((ext_vector_type(2))) float v2f;
typedef __attribute__((ext_vector_type(8))) float v8f;

#define DIM 64

// ---------------------------------------------------------------------------
// 1) node init: deg = 1 (self loop), gsum = 0, gcnt = 0
__global__ __launch_bounds__(256) void k_init_nodes(float* __restrict__ deg,
                                                    float* __restrict__ gsum,
                                                    float* __restrict__ gcnt, int n) {
  int i = blockIdx.x * blockDim.x + threadIdx.x;
  if (i < n) { deg[i] = 1.0f; gsum[i] = 0.0f; gcnt[i] = 0.0f; }
}

// 2) in-degree accumulation over edge dst
__global__ __launch_bounds__(256) void k_deg(const int* __restrict__ dst,
                                             float* __restrict__ deg, int e) {
  int i = blockIdx.x * blockDim.x + threadIdx.x;
  if (i < e) atomicAdd(&deg[dst[i]], 1.0f);
}

// 3) deg -> rsqrt(deg) in place
__global__ __launch_bounds__(256) void k_rsqrt(float* __restrict__ deg, int n) {
  int i = blockIdx.x * blockDim.x + threadIdx.x;
  if (i < n) deg[i] = rsqrtf(deg[i]);
}

// 4) h = X @ W via V_WMMA_F32_16X16X4_F32.
//    Block = 128 threads = 4 waves; wave w computes the 16x16 tile at
//    (m0 = 16*blockIdx.x, n0 = 16*w). K stepped by 4, W staged in LDS.
__global__ __launch_bounds__(128) void k_gemm_wmma(const float* __restrict__ X,
                                                   const float* __restrict__ Wm,
                                                   float* __restrict__ h, int n) {
  __shared__ float Ws[DIM * DIM];
  for (int i = threadIdx.x; i < DIM * DIM; i += 128) Ws[i] = Wm[i];
  __syncthreads();

  const int lane = threadIdx.x & 31;
  const int wave = threadIdx.x >> 5;
  const int m0 = blockIdx.x * 16;
  const int n0 = wave * 16;
  const int nn = lane & 15;       // column within tile / row-in-half for A
  const int hi = lane >> 4;       // lane half: 0 or 1
  const int kh = hi * 2;          // A/B ISA layout: lanes>=16 hold K+2,K+3

  int arow = m0 + nn;
  if (arow >= n) arow = n - 1;    // clamp loads; stores are guarded below
  const float* Xr = X + (size_t)arow * DIM;

  v8f c = {};
#pragma unroll
  for (int k0 = 0; k0 < DIM; k0 += 4) {
    v2f a, b;
    a.x = Xr[k0 + kh + 0];
    a.y = Xr[k0 + kh + 1];
    b.x = Ws[(k0 + kh + 0) * DIM + n0 + nn];
    b.y = Ws[(k0 + kh + 1) * DIM + n0 + nn];
    // 8-arg form: (neg_a, A, neg_b, B, c_mod, C, reuse_a, reuse_b)
    c = __builtin_amdgcn_wmma_f32_16x16x4_f32(false, a, false, b, (short)0, c,
                                              false, false);
  }

  // C/D layout: VGPR j -> M = j + 8*hi, N = lane%16
  const int mbase = m0 + hi * 8;
#pragma unroll
  for (int j = 0; j < 8; ++j) {
    int r = mbase + j;
    if (r < n) h[(size_t)r * DIM + n0 + nn] = c[j];
  }
}

// 5) seed agg with self-loop term: agg[i][d] = h[i][d] * dinv[i]^2
__global__ __launch_bounds__(256) void k_self(const float* __restrict__ h,
                                              const float* __restrict__ dinv,
                                              float* __restrict__ agg, int total) {
  int t = blockIdx.x * blockDim.x + threadIdx.x;
  if (t < total) {
    float dv = dinv[t >> 6];
    agg[t] = h[t] * dv * dv;
  }
}

// 6) edge aggregation: 64 threads per edge, one dim each, L2-resident atomics
__global__ __launch_bounds__(256) void k_edge_agg(const int* __restrict__ src,
                                                  const int* __restrict__ dst,
                                                  const float* __restrict__ h,
                                                  const float* __restrict__ dinv,
                                                  float* __restrict__ agg, int e) {
  int t = blockIdx.x * blockDim.x + threadIdx.x;
  int ed = t >> 6;
  if (ed >= e) return;
  int d = t & 63;
  int s = src[ed], q = dst[ed];
  float norm = dinv[s] * dinv[q];
  atomicAdd(&agg[(size_t)q * DIM + d], h[(size_t)s * DIM + d] * norm);
}

// 7) H = relu(agg + b), in place
__global__ __launch_bounds__(256) void k_relu(float* __restrict__ agg,
                                              const float* __restrict__ bias, int total) {
  int t = blockIdx.x * blockDim.x + threadIdx.x;
  if (t < total) agg[t] = fmaxf(agg[t] + bias[t & 63], 0.0f);
}

// 8) G2 gate: 16 lanes per edge, float4 loads, shfl-xor reduction (wave32,
//    xor masks <= 8 stay inside each aligned 16-lane group)
__global__ __launch_bounds__(256) void k_gate(const int* __restrict__ src,
                                              const int* __restrict__ dst,
                                              const float* __restrict__ H,
                                              float* __restrict__ gsum,
                                              float* __restrict__ gcnt, int e) {
  int t = blockIdx.x * blockDim.x + threadIdx.x;
  int ed = t >> 4;
  if (ed >= e) return;
  int l = t & 15;
  int r = src[ed], c = dst[ed];
  const float4* Hr = (const float4*)(H + (size_t)r * DIM);
  const float4* Hc = (const float4*)(H + (size_t)c * DIM);
  float4 a = Hr[l], b = Hc[l];
  float dx = a.x - b.x, dy = a.y - b.y, dz = a.z - b.z, dw = a.w - b.w;
  float s = dx * dx + dy * dy + dz * dz + dw * dw;  // |.|^2 == P=2 path
  s += __shfl_xor(s, 8);
  s += __shfl_xor(s, 4);
  s += __shfl_xor(s, 2);
  s += __shfl_xor(s, 1);
  if (l == 0) {
    atomicAdd(&gsum[r], s);
    atomicAdd(&gcnt[r], 1.0f);
  }
}

// 9) out = tanh(gsum / max(cnt, 1))
__global__ __launch_bounds__(256) void k_final(const float* __restrict__ gsum,
                                               const float* __restrict__ gcnt,
                                               float* __restrict__ out, int n) {
  int i = blockIdx.x * blockDim.x + threadIdx.x;
  if (i < n) out[i] = tanhf(gsum[i] / fmaxf(gcnt[i], 1.0f));
}

extern "C" void kernel_launch(void* const* d_in, const int* in_sizes, int n_in,
                              void* d_out, int out_size, void* d_ws, size_t ws_size,
                              hipStream_t stream) {
  const int N = in_sizes[0] / DIM;
  const int E = in_sizes[1] / 2;
  const float* X  = (const float*)d_in[0];
  const int*   ei = (const int*)d_in[1];
  const int*   src = ei;
  const int*   dst = ei + E;
  const float* W  = (const float*)d_in[2];
  const float* b  = (const float*)d_in[3];
  float* out = (float*)d_out;

  // workspace carve-out (256B aligned slices)
  char* ws = (char*)d_ws;
  size_t off = 0;
  auto carve = [&](size_t bytes) -> float* {
    float* p = (float*)(ws + off);
    off = (off + bytes + 255) & ~(size_t)255;
    return p;
  };
  float* dinv = carve((size_t)N * 4);          // deg, then rsqrt in place
  float* h    = carve((size_t)N * DIM * 4);    // X @ W
  float* agg  = carve((size_t)N * DIM * 4);    // GCN accum -> H in place
  float* gsum = carve((size_t)N * 4);
  float* gcnt = carve((size_t)N * 4);

  const int total = N * DIM;

  k_init_nodes<<<(N + 255) / 256, 256, 0, stream>>>(dinv, gsum, gcnt, N);
  k_deg<<<(E + 255) / 256, 256, 0, stream>>>(dst, dinv, E);
  k_rsqrt<<<(N + 255) / 256, 256, 0, stream>>>(dinv, N);
  k_gemm_wmma<<<(N + 15) / 16, 128, 0, stream>>>(X, W, h, N);
  k_self<<<(total + 255) / 256, 256, 0, stream>>>(h, dinv, agg, total);
  {
    long long thr = (long long)E * 64;
    k_edge_agg<<<(int)((thr + 255) / 256), 256, 0, stream>>>(src, dst, h, dinv, agg, E);
  }
  k_relu<<<(total + 255) / 256, 256, 0, stream>>>(agg, b, total);
  {
    long long thr = (long long)E * 16;
    k_gate<<<(int)((thr + 255) / 256), 256, 0, stream>>>(src, dst, agg, gsum, gcnt, E);
  }
  k_final<<<(N + 255) / 256, 256, 0, stream>>>(gsum, gcnt, out, N);
}